// GTN_42614665511413
// MI455X (gfx1250) — compile-verified
//
#include <hip/hip_runtime.h>

typedef float v2f __attribute__((ext_vector_type(2)));
typedef float v8f __attribute__((ext_vector_type(8)));

constexpr int    kN  = 2048;
constexpr size_t kNN = (size_t)kN * kN;
constexpr int    kC  = 2;
constexpr int    kE  = 5;

// ---------------- tiny: 3 softmaxes over [C,E] ----------------
__global__ void gtn_softmax3(const float* w0, const float* w1, const float* w2,
                             float* s_ws, float* s_out) {
    if (threadIdx.x != 0 || blockIdx.x != 0) return;
    const float* ws_[3] = {w0, w1, w2};
    for (int t = 0; t < 3; ++t) {
        for (int c = 0; c < kC; ++c) {
            float m = -1e30f;
            for (int e = 0; e < kE; ++e) m = fmaxf(m, ws_[t][c * kE + e]);
            float tmp[kE]; float sum = 0.f;
            for (int e = 0; e < kE; ++e) { tmp[e] = expf(ws_[t][c * kE + e] - m); sum += tmp[e]; }
            for (int e = 0; e < kE; ++e) {
                float v = tmp[e] / sum;
                s_ws[t * kC * kE + c * kE + e]  = v;
                s_out[t * kC * kE + c * kE + e] = v;
            }
        }
    }
}

// ---------------- comb[c] = sum_e s[c,e] * A[e]  (1 or 2 outputs), float4 streams ----------------
__global__ void gtn_comb(const float4* __restrict__ A4,
                         const float* __restrict__ sA, const float* __restrict__ sB,
                         float4* __restrict__ outA4, float4* __restrict__ outB4) {
    const size_t total4 = kNN / 4;
    size_t idx    = (size_t)blockIdx.x * blockDim.x + threadIdx.x;
    size_t stride = (size_t)gridDim.x * blockDim.x;
    for (size_t i = idx; i < total4; i += stride) {
        float4 a[kE];
#pragma unroll
        for (int e = 0; e < kE; ++e) a[e] = A4[(size_t)e * total4 + i];
#pragma unroll
        for (int c = 0; c < kC; ++c) {
            float4 h1 = {0.f, 0.f, 0.f, 0.f};
#pragma unroll
            for (int e = 0; e < kE; ++e) {
                const float s = sA[c * kE + e];
                h1.x = fmaf(s, a[e].x, h1.x); h1.y = fmaf(s, a[e].y, h1.y);
                h1.z = fmaf(s, a[e].z, h1.z); h1.w = fmaf(s, a[e].w, h1.w);
            }
            outA4[(size_t)c * total4 + i] = h1;
            if (outB4) {
                float4 h2 = {0.f, 0.f, 0.f, 0.f};
#pragma unroll
                for (int e = 0; e < kE; ++e) {
                    const float s = sB[c * kE + e];
                    h2.x = fmaf(s, a[e].x, h2.x); h2.y = fmaf(s, a[e].y, h2.y);
                    h2.z = fmaf(s, a[e].z, h2.z); h2.w = fmaf(s, a[e].w, h2.w);
                }
                outB4[(size_t)c * total4 + i] = h2;
            }
        }
    }
}

// ---------------- fp32 WMMA GEMM: D[c] = A[c] @ B[c], 2048^3 per channel ----------------
// 128x128 block tile, 8 waves (2x4), 64x32 per wave; BK=32 K-slabs.
// LDS ping-pong double buffer fed by GLOBAL_LOAD_ASYNC_TO_LDS_B128 (ASYNCcnt):
// slab k+1 is DMA'd into the alternate buffer while the 64 WMMAs of slab k run.
constexpr int BM = 128, BN = 128, BK = 32;
constexpr int ALD = 36;   // padded LDS row stride (floats) for A tile: 144B -> float4-aligned
constexpr int BLD = 132;  // padded LDS row stride (floats) for B tile: 528B -> float4-aligned

__global__ __launch_bounds__(256)
void gtn_wmma_gemm(const float* __restrict__ Asrc, const float* __restrict__ Bsrc,
                   float* __restrict__ Dst) {
    __shared__ __align__(16) float Alds[2][BM * ALD];  // 2 x 18.0 KB
    __shared__ __align__(16) float Blds[2][BK * BLD];  // 2 x 16.5 KB

    const int c = blockIdx.z;
    const size_t base = (size_t)c * kNN;
    const float* Ap = Asrc + base;
    const float* Bp = Bsrc + base;
    float*       Dp = Dst  + base;

    const int t     = threadIdx.x;
    const int wave  = t >> 5;
    const int lane  = t & 31;
    const int lm    = lane & 15;   // row (A) / col (B,D) within 16x16 tile
    const int lk    = lane >> 4;   // half-wave select
    const int waveM = (wave & 1) * 64;   // 2 waves along M
    const int waveN = (wave >> 1) * 32;  // 4 waves along N
    const int blockM = blockIdx.y * BM;
    const int blockN = blockIdx.x * BN;

    // async DMA of one 128x32 A slab + 32x128 B slab into LDS buffer `buf`
    auto asyncLoadSlab = [&](int k0, int buf) {
#pragma unroll
        for (int p = 0; p < 4; ++p) {
            const int idx = p * 256 + t;
            const float* ga = Ap + (size_t)(blockM + (idx >> 3)) * kN + k0 + (idx & 7) * 4;
            unsigned la = (unsigned)(uintptr_t)&Alds[buf][(idx >> 3) * ALD + (idx & 7) * 4];
            asm volatile("global_load_async_to_lds_b128 %0, %1, off"
                         :: "v"(la), "v"(ga) : "memory");
            const float* gb = Bp + (size_t)(k0 + (idx >> 5)) * kN + blockN + (idx & 31) * 4;
            unsigned lb = (unsigned)(uintptr_t)&Blds[buf][(idx >> 5) * BLD + (idx & 31) * 4];
            asm volatile("global_load_async_to_lds_b128 %0, %1, off"
                         :: "v"(lb), "v"(gb) : "memory");
        }
    };

    v8f acc[4][2];
#pragma unroll
    for (int i = 0; i < 4; ++i)
#pragma unroll
        for (int j = 0; j < 2; ++j)
            acc[i][j] = (v8f){0.f, 0.f, 0.f, 0.f, 0.f, 0.f, 0.f, 0.f};

    // prologue: DMA slab 0 into buffer 0
    asyncLoadSlab(0, 0);
    asm volatile("s_wait_asynccnt 0x0" ::: "memory");
    __syncthreads();

    int cur = 0;
    for (int k0 = 0; k0 < kN; k0 += BK) {
        const bool hasNext = (k0 + BK) < kN;
        if (hasNext) asyncLoadSlab(k0 + BK, cur ^ 1);  // DMA in flight during WMMAs

        const float* __restrict__ Ac = Alds[cur];
        const float* __restrict__ Bc = Blds[cur];
#pragma unroll
        for (int ks = 0; ks < BK / 4; ++ks) {
            const int kk = ks * 4 + 2 * lk;  // this lane's K pair within the slab
            v2f afr[4];
#pragma unroll
            for (int tm = 0; tm < 4; ++tm)
                afr[tm] = *(const v2f*)(&Ac[(waveM + tm * 16 + lm) * ALD + kk]);
            v2f bfr[2];
#pragma unroll
            for (int tn = 0; tn < 2; ++tn) {
                const float* bp = &Bc[kk * BLD + waveN + tn * 16 + lm];
                bfr[tn].x = bp[0];
                bfr[tn].y = bp[BLD];
            }
#pragma unroll
            for (int tm = 0; tm < 4; ++tm)
#pragma unroll
                for (int tn = 0; tn < 2; ++tn)
                    acc[tm][tn] = __builtin_amdgcn_wmma_f32_16x16x4_f32(
                        false, afr[tm], false, bfr[tn],
                        (short)0, acc[tm][tn], false, false);
        }

        asm volatile("s_wait_asynccnt 0x0" ::: "memory");  // this wave's DMA landed
        __syncthreads();                                   // all waves' DMA visible
        cur ^= 1;
    }

    // epilogue: D layout -> VGPR r holds (M = r + 8*lk, N = lm) within each 16x16 tile
#pragma unroll
    for (int tm = 0; tm < 4; ++tm) {
#pragma unroll
        for (int tn = 0; tn < 2; ++tn) {
            const int colg = blockN + waveN + tn * 16 + lm;
#pragma unroll
            for (int r = 0; r < 8; ++r) {
                const int rowg = blockM + waveM + tm * 16 + r + 8 * lk;
                Dp[(size_t)rowg * kN + colg] = acc[tm][tn][r];
            }
        }
    }
}

// ---------------- row sums -> dinv = deg^-1/2 (0 if deg == 0) ----------------
__global__ void gtn_rowinv(const float* __restrict__ T, float* __restrict__ dinv) {
    __shared__ float red[256];
    const int r = blockIdx.x;               // 0 .. C*N-1
    const float4* row4 = (const float4*)(T + (size_t)r * kN);
    float s = 0.f;
    for (int j = threadIdx.x; j < kN / 4; j += 256) {
        float4 v = row4[j];
        s += (v.x + v.y) + (v.z + v.w);
    }
    red[threadIdx.x] = s;
    __syncthreads();
    for (int o = 128; o > 0; o >>= 1) {
        if (threadIdx.x < o) red[threadIdx.x] += red[threadIdx.x + o];
        __syncthreads();
    }
    if (threadIdx.x == 0) {
        float d = red[0];
        dinv[r] = (d > 0.f) ? rsqrtf(d) : 0.f;
    }
}

// ---------------- H = dinv[row] * T (src may equal dst), float4 streams ----------------
__global__ void gtn_scale(const float4* __restrict__ src4, const float* __restrict__ dinv,
                          float4* __restrict__ dst4) {
    const size_t total4 = (size_t)kC * kNN / 4;
    size_t idx    = (size_t)blockIdx.x * blockDim.x + threadIdx.x;
    size_t stride = (size_t)gridDim.x * blockDim.x;
    for (size_t i = idx; i < total4; i += stride) {
        const float d = dinv[i >> 9];       // 512 float4 per row; i>>9 == c*N + row
        float4 v = src4[i];
        v.x *= d; v.y *= d; v.z *= d; v.w *= d;
        dst4[i] = v;
    }
}

// ---------------- head: y = relu(X@Wg.T+bg)[tx] @ folded(lin_w).T + lin_b ----------------
__global__ void gtn_head(const float* __restrict__ X, const float* __restrict__ gw,
                         const float* __restrict__ gb, const float* __restrict__ lw,
                         const float* __restrict__ lb, const int* __restrict__ tx,
                         float* __restrict__ y) {
    __shared__ float h[128];
    const int t   = blockIdx.x;             // 0..511 targets
    const int j   = threadIdx.x;            // 128 hidden units
    const int row = tx[t];
    const float4* xr4 = (const float4*)(X  + (size_t)row * 512);
    const float4* wr4 = (const float4*)(gw + (size_t)j   * 512);
    float acc = 0.f;
#pragma unroll 4
    for (int k = 0; k < 128; ++k) {
        float4 a = xr4[k], b = wr4[k];
        acc = fmaf(a.x, b.x, acc); acc = fmaf(a.y, b.y, acc);
        acc = fmaf(a.z, b.z, acc); acc = fmaf(a.w, b.w, acc);
    }
    h[j] = fmaxf(acc + gb[j], 0.f);
    __syncthreads();
    if (j < 8) {
        float a = lb[j];
        const float* l = lw + j * 256;      // concat([x]*C) folds: lw[j][q] + lw[j][128+q]
        for (int q = 0; q < 128; ++q) a = fmaf(h[q], l[q] + l[128 + q], a);
        y[t * 8 + j] = a;
    }
}

extern "C" void kernel_launch(void* const* d_in, const int* in_sizes, int n_in,
                              void* d_out, int out_size, void* d_ws, size_t ws_size,
                              hipStream_t stream) {
    const float* A     = (const float*)d_in[0];
    const float* X     = (const float*)d_in[1];
    const float* w0_1  = (const float*)d_in[2];
    const float* w0_2  = (const float*)d_in[3];
    const float* w1_1  = (const float*)d_in[4];
    const float* gcn_w = (const float*)d_in[5];
    const float* gcn_b = (const float*)d_in[6];
    const float* lin_w = (const float*)d_in[7];
    const float* lin_b = (const float*)d_in[8];
    const int*   tx    = (const int*)d_in[9];

    float* out    = (float*)d_out;
    float* y_out  = out;                    // 512*8 = 4096
    float* ws_out = out + 4096;             // 3*C*E = 30
    float* h_out  = out + 4096 + 30;        // C*N*N

    const size_t BUF = (size_t)kC * kNN;    // 8,388,608 floats per buffer
    float* bufA = (float*)d_ws;             // HA, later HB2
    float* bufB = bufA + BUF;               // HB, later T2
    float* bufT = bufB + BUF;               // T1 -> H1 (in place)
    float* dinv = bufT + BUF;               // C*N
    float* sbuf = dinv + (size_t)kC * kN;   // 30 softmax values

    dim3 gemmGrid(kN / BN, kN / BM, kC);

    gtn_softmax3<<<1, 32, 0, stream>>>(w0_1, w0_2, w1_1, sbuf, ws_out);
    // layer 0: HA = comb(s1, A), HB = comb(s2, A); T1 = HA @ HB; H1 = norm(T1)
    gtn_comb<<<2048, 256, 0, stream>>>((const float4*)A, sbuf + 0, sbuf + 10,
                                       (float4*)bufA, (float4*)bufB);
    gtn_wmma_gemm<<<gemmGrid, 256, 0, stream>>>(bufA, bufB, bufT);
    gtn_rowinv<<<kC * kN, 256, 0, stream>>>(bufT, dinv);
    gtn_scale<<<4096, 256, 0, stream>>>((const float4*)bufT, dinv, (float4*)bufT);
    // layer 1: HB2 = comb(s3, A); T2 = H1 @ HB2; H = norm(T2) -> d_out
    gtn_comb<<<2048, 256, 0, stream>>>((const float4*)A, sbuf + 20, nullptr,
                                       (float4*)bufA, nullptr);
    gtn_wmma_gemm<<<gemmGrid, 256, 0, stream>>>(bufT, bufA, bufB);
    gtn_rowinv<<<kC * kN, 256, 0, stream>>>(bufB, dinv);
    gtn_scale<<<4096, 256, 0, stream>>>((const float4*)bufB, dinv, (float4*)h_out);
    // classification head
    gtn_head<<<512, 128, 0, stream>>>(X, gcn_w, gcn_b, lin_w, lin_b, tx, y_out);
}